// SAGELayer_68195490726430
// MI455X (gfx1250) — compile-verified
//
#include <hip/hip_runtime.h>

// GraphSAGE layer for MI455X (gfx1250, wave32).
//   out = (segsum(x[src] by dst)/max(cnt,1)) @ W_l^T + b_l + x @ W_r^T
// Phase A: zero + per-edge scatter with global_atomic_add_f32 (L2-resident 25.6MB acc),
//          one wave per edge, one global_load_b128 per lane.
// Phase B: fused GEMM  [agg_mean | x] (50000x256) @ [W_l^T ; W_r^T] (256x128)
//          via V_WMMA_F32_16X16X4_F32, one 16x16 tile per wave, split-K loop
//          (no per-iteration pointer selects).

typedef __attribute__((ext_vector_type(2))) float v2f;
typedef __attribute__((ext_vector_type(8))) float v8f;

#define N_NODES 50000
#define N_EDGES 800000
#define D 128
#define AS_STRIDE 260   // floats; 260 % 64 == 4 -> conflict-free half-wave frag reads, 8B aligned

__global__ void sage_zero(float* __restrict__ ws, int n) {
  int i = blockIdx.x * blockDim.x + threadIdx.x;
  int stride = gridDim.x * blockDim.x;
  for (; i < n; i += stride) ws[i] = 0.0f;
}

__global__ void sage_scatter(const float* __restrict__ x,
                             const int* __restrict__ src,
                             const int* __restrict__ dst,
                             float* __restrict__ agg,
                             float* __restrict__ cnt) {
  int e = (int)((blockIdx.x * blockDim.x + threadIdx.x) >> 5);  // one wave per edge
  int lane = threadIdx.x & 31;
  if (e >= N_EDGES) return;
  int s = src[e];
  int d = dst[e];
  // lane owns 4 consecutive floats: one b128 load, 4 dword atomics (512B/wave).
  const float4* xr = (const float4*)(x + (size_t)s * D);  // 16B aligned: s*512
  float4 v = xr[lane];
  float* ar = agg + (size_t)d * D + lane * 4;
  atomicAdd(ar + 0, v.x);
  atomicAdd(ar + 1, v.y);
  atomicAdd(ar + 2, v.z);
  atomicAdd(ar + 3, v.w);
  if (lane == 0) atomicAdd(&cnt[d], 1.0f);
}

__global__ __launch_bounds__(256) void sage_gemm(const float* __restrict__ x,
                                                 const float* __restrict__ Wl,
                                                 const float* __restrict__ bl,
                                                 const float* __restrict__ Wr,
                                                 const float* __restrict__ agg,
                                                 const float* __restrict__ cnt,
                                                 float* __restrict__ out) {
  __shared__ float As[16 * AS_STRIDE];

  const int tid = threadIdx.x;
  const int node0 = blockIdx.x * 16;  // 3125 * 16 == 50000 exactly, no tail

  // Stage A-panel: rows = 16 nodes, cols = 256 (mean-scaled agg | x). Coalesced.
#pragma unroll
  for (int j = 0; j < 16; ++j) {
    int node = node0 + j;
    int col = tid;  // 0..255
    float v;
    if (col < 128) {
      float c = cnt[node];
      v = agg[(size_t)node * D + col] / fmaxf(c, 1.0f);
    } else {
      v = x[(size_t)node * D + (col - 128)];
    }
    As[j * AS_STRIDE + col] = v;
  }
  __syncthreads();

  const int lane = tid & 31;
  const int wave = tid >> 5;     // 8 waves -> 8 N-tiles of 16 (covers D_OUT=128)
  const int n0 = wave * 16;
  const int hi = lane >> 4;      // half-wave: 0 -> K,K+1 ; 1 -> K+2,K+3
  const int lm = lane & 15;
  const int n = n0 + lm;         // output column owned by this lane in B-frag

  const float* Wln = Wl + (size_t)n * D;   // B[k][n]     = W_l[n][k]
  const float* Wrn = Wr + (size_t)n * D;   // B[128+k][n] = W_r[n][k]
  const float* Arow = &As[lm * AS_STRIDE];
  const int koff = 2 * hi;       // per-ISA 16x4 f32 fragment split across half-waves

  v8f acc = {};
  // First half of K: A = agg_mean panel, B = W_l. All 8B accesses aligned.
#pragma unroll 8
  for (int k = 0; k < 128; k += 4) {
    const int kk = k + koff;
    v2f a = *(const v2f*)(Arow + kk);
    v2f b = *(const v2f*)(Wln + kk);
    acc = __builtin_amdgcn_wmma_f32_16x16x4_f32(false, a, false, b, (short)0, acc,
                                                false, false);
  }
  // Second half of K: A = x panel, B = W_r.
#pragma unroll 8
  for (int k = 0; k < 128; k += 4) {
    const int kk = k + koff;
    v2f a = *(const v2f*)(Arow + 128 + kk);
    v2f b = *(const v2f*)(Wrn + kk);
    acc = __builtin_amdgcn_wmma_f32_16x16x4_f32(false, a, false, b, (short)0, acc,
                                                false, false);
  }

  // C/D layout: VGPR r -> M = r + 8*hi, N = n. Add bias, store.
  const float bias = bl[n];
#pragma unroll
  for (int r = 0; r < 8; ++r) {
    int m = r + 8 * hi;
    out[(size_t)(node0 + m) * D + n] = acc[r] + bias;
  }
}

extern "C" void kernel_launch(void* const* d_in, const int* in_sizes, int n_in,
                              void* d_out, int out_size, void* d_ws, size_t ws_size,
                              hipStream_t stream) {
  const float* x  = (const float*)d_in[0];
  const int* ei   = (const int*)d_in[1];      // [2, E]: src row then dst row
  const float* Wl = (const float*)d_in[2];
  const float* bl = (const float*)d_in[3];
  const float* Wr = (const float*)d_in[4];
  float* out = (float*)d_out;

  const int* src = ei;
  const int* dst = ei + N_EDGES;

  float* agg = (float*)d_ws;                       // N_NODES * D floats
  float* cnt = agg + (size_t)N_NODES * D;          // N_NODES floats

  const int nzero = N_NODES * D + N_NODES;
  sage_zero<<<2048, 256, 0, stream>>>((float*)d_ws, nzero);

  const int edges_per_block = 256 / 32;            // one wave per edge
  sage_scatter<<<(N_EDGES + edges_per_block - 1) / edges_per_block, 256, 0, stream>>>(
      x, src, dst, agg, cnt);

  sage_gemm<<<N_NODES / 16, 256, 0, stream>>>(x, Wl, bl, Wr, agg, cnt, out);
}